// MoEGate_54546084659699
// MI455X (gfx1250) — compile-verified
//
#include <hip/hip_runtime.h>
#include <hip/hip_bf16.h>
#include <math.h>

// Problem constants (match reference: B=4, S=4096, D=4096, E=16, TOP_K=2)
#define NTOK   16384
#define DDIM   4096
#define NEXP   16
#define WAVES_PER_BLOCK 4
#define TOK_PER_WAVE    16

typedef __attribute__((ext_vector_type(2))) float v2f;
typedef __attribute__((ext_vector_type(8))) float v8f;

__device__ __forceinline__ v2f ld_nt(const float* p) {
  return __builtin_nontemporal_load((const v2f*)p);
}
__device__ __forceinline__ v2f ld(const float* p) {
  return *(const v2f*)p;
}

// ---------------------------------------------------------------------------
// Zero the workspace accumulators (graph-capture-safe init each call).
// ws layout: [0..15] float usage sums, [16..31] u32 top-k counts
// ---------------------------------------------------------------------------
__global__ void moe_gate_zero(float* g_usage, unsigned* g_counts) {
  int t = threadIdx.x;
  if (t < NEXP) { g_usage[t] = 0.0f; g_counts[t] = 0u; }
}

// ---------------------------------------------------------------------------
// Main kernel: one wave32 per 16-token tile.
//   gate/noise logits via V_WMMA_F32_16X16X4_F32 accumulated over D=4096.
//   Fragments for 4 K-steps (16 floats of K) are batched into arrays (12
//   independent b64 loads per chunk) and two chunks are ping-ponged so
//   ~12-24 loads stay in flight per wave while WMMAs consume the other
//   chunk. x is streamed non-temporal; W_gate/W_noise stay temporal (L2).
// A-frag (16x4 f32): lane L -> row L&15, float2 at k0 + 2*(L>=16)
// B-frag (4x16 f32): lane L -> expert-row L&15, float2 at k0 + 2*(L>=16)
// C/D   (16x16 f32): VGPR j, lane L -> token (j + 8*(L>=16)), expert L&15
// ---------------------------------------------------------------------------
__global__ __launch_bounds__(WAVES_PER_BLOCK * 32) void moe_gate_main(
    const float* __restrict__ x,      // [NTOK, DDIM]
    const float* __restrict__ Wg,     // [NEXP, DDIM]
    const float* __restrict__ Wn,     // [NEXP, DDIM]
    const float* __restrict__ noise,  // [NTOK, NEXP]
    float* __restrict__ out_gw,       // [NTOK, NEXP]
    float* __restrict__ out_idx,      // [NTOK, 2] (as float)
    float* __restrict__ out_tw,       // [NTOK, 2]
    float* __restrict__ g_usage,      // [NEXP]
    unsigned* __restrict__ g_counts)  // [NEXP]
{
  __shared__ float    s_usage[NEXP];
  __shared__ unsigned s_counts[NEXP];
  const int tid = threadIdx.x;
  if (tid < NEXP) { s_usage[tid] = 0.0f; s_counts[tid] = 0u; }
  __syncthreads();

  const int lane = tid & 31;
  const int wave = tid >> 5;
  const int tile = blockIdx.x * WAVES_PER_BLOCK + wave;
  const int hi   = lane >> 4;   // 0 = lanes 0-15, 1 = lanes 16-31
  const int e    = lane & 15;   // expert index (and A/B fragment row)
  const int t0   = tile * TOK_PER_WAVE;

  const float* pa = x  + (size_t)(t0 + e) * DDIM + 2 * hi;
  const float* pg = Wg + (size_t)e        * DDIM + 2 * hi;
  const float* pn = Wn + (size_t)e        * DDIM + 2 * hi;

  v8f accg = {};
  v8f accn = {};

  // Ping-pong staging: each chunk = 4 K-steps = 16 floats of K.
  v2f A0[4], G0[4], N0[4];
  v2f A1[4], G1[4], N1[4];

#pragma unroll
  for (int i = 0; i < 4; ++i) {          // preload chunk at k=0
    A0[i] = ld_nt(pa + 4 * i);
    G0[i] = ld(pg + 4 * i);
    N0[i] = ld(pn + 4 * i);
  }

#pragma unroll 1
  for (int k = 0; k < DDIM; k += 32) {
    // prefetch chunk at k+16 (always in-bounds within this iteration)
#pragma unroll
    for (int i = 0; i < 4; ++i) {
      A1[i] = ld_nt(pa + k + 16 + 4 * i);
      G1[i] = ld(pg + k + 16 + 4 * i);
      N1[i] = ld(pn + k + 16 + 4 * i);
    }
    // consume chunk at k
#pragma unroll
    for (int i = 0; i < 4; ++i) {
      accg = __builtin_amdgcn_wmma_f32_16x16x4_f32(false, A0[i], false, G0[i],
                                                   (short)0, accg, false, false);
      accn = __builtin_amdgcn_wmma_f32_16x16x4_f32(false, A0[i], false, N0[i],
                                                   (short)0, accn, false, false);
    }
    // prefetch chunk at k+32 (wraps to 0 on the last iteration; benign reload)
    const int kp = (k + 32) & (DDIM - 1);
#pragma unroll
    for (int i = 0; i < 4; ++i) {
      A0[i] = ld_nt(pa + kp + 4 * i);
      G0[i] = ld(pg + kp + 4 * i);
      N0[i] = ld(pn + kp + 4 * i);
    }
    // consume chunk at k+16
#pragma unroll
    for (int i = 0; i < 4; ++i) {
      accg = __builtin_amdgcn_wmma_f32_16x16x4_f32(false, A1[i], false, G1[i],
                                                   (short)0, accg, false, false);
      accn = __builtin_amdgcn_wmma_f32_16x16x4_f32(false, A1[i], false, N1[i],
                                                   (short)0, accn, false, false);
    }
  }

  float usage_acc = 0.0f;
#pragma unroll
  for (int j = 0; j < 8; ++j) {
    const int token = t0 + j + 8 * hi;

    // noisy gating: logit += noise * softplus(noise_logit) * 0.01
    float nl = accn[j];
    float sp = (nl > 20.0f) ? nl : log1pf(expf(nl));
    float nz = __builtin_nontemporal_load(&noise[(size_t)token * NEXP + e]);
    float logit = accg[j] + nz * sp * 0.01f;

    // softmax over 16 experts (butterfly inside each 16-lane group)
    float mx = logit;
#pragma unroll
    for (int m = 1; m < 16; m <<= 1) mx = fmaxf(mx, __shfl_xor(mx, m));
    float ex = expf(logit - mx);
    float sm = ex;
#pragma unroll
    for (int m = 1; m < 16; m <<= 1) sm += __shfl_xor(sm, m);
    float w = ex / sm;

    __builtin_nontemporal_store(w, &out_gw[(size_t)token * NEXP + e]);
    usage_acc += w;

    // top-1 (argmax, ties -> lowest index like lax.top_k)
    float b1v = w; int b1i = e;
#pragma unroll
    for (int m = 1; m < 16; m <<= 1) {
      float ov = __shfl_xor(b1v, m);
      int   oi = __shfl_xor(b1i, m);
      if (ov > b1v || (ov == b1v && oi < b1i)) { b1v = ov; b1i = oi; }
    }
    // top-2: mask out winner (weights are in (0,1), -1 is a safe sentinel)
    float b2v = (e == b1i) ? -1.0f : w; int b2i = e;
#pragma unroll
    for (int m = 1; m < 16; m <<= 1) {
      float ov = __shfl_xor(b2v, m);
      int   oi = __shfl_xor(b2i, m);
      if (ov > b2v || (ov == b2v && oi < b2i)) { b2v = ov; b2i = oi; }
    }

    if (e == 0) {
      float s = b1v + b2v + 1e-8f;
      __builtin_nontemporal_store((float)b1i, &out_idx[token * 2 + 0]);
      __builtin_nontemporal_store((float)b2i, &out_idx[token * 2 + 1]);
      __builtin_nontemporal_store(b1v / s,    &out_tw[token * 2 + 0]);
      __builtin_nontemporal_store(b2v / s,    &out_tw[token * 2 + 1]);
      atomicAdd(&s_counts[b1i], 1u);
      atomicAdd(&s_counts[b2i], 1u);
    }
  }

  atomicAdd(&s_usage[e], usage_acc);
  __syncthreads();
  if (tid < NEXP) {
    atomicAdd(&g_usage[tid], s_usage[tid]);
    atomicAdd(&g_counts[tid], s_counts[tid]);
  }
}

// ---------------------------------------------------------------------------
// Finalize: 5 scalar stats from the 16-expert accumulators (trivial work).
// ---------------------------------------------------------------------------
__global__ void moe_gate_finalize(const float* __restrict__ g_usage,
                                  const unsigned* __restrict__ g_counts,
                                  float* __restrict__ out5) {
  if (threadIdx.x != 0) return;
  const float Nf  = (float)NTOK;
  const float cap = Nf * 1.25f / (float)NEXP;

  float lbl = 0.0f, total = 0.0f, umean = 0.0f;
  float util[NEXP];
  for (int i = 0; i < NEXP; ++i) {
    float d = g_usage[i] / Nf - 1.0f / (float)NEXP;
    lbl += d * d;
    float c = (float)g_counts[i];
    total += c;
    util[i] = c / cap;
    umean += util[i];
  }
  lbl   /= (float)NEXP;
  umean /= (float)NEXP;

  float var = 0.0f, ent = 0.0f, ovf = 0.0f;
  for (int i = 0; i < NEXP; ++i) {
    float d = util[i] - umean;
    var += d * d;
    float p = (float)g_counts[i] / total + 1e-8f;
    ent -= p * logf(p);
    if (util[i] > 1.0f) ovf += 1.0f;
  }
  out5[0] = lbl;                               // load_balancing_loss
  out5[1] = umean;                             // cap_mean
  out5[2] = sqrtf(var / (float)(NEXP - 1));    // cap_std (unbiased)
  out5[3] = ent;                               // entropy
  out5[4] = ovf / (float)NEXP;                 // overflow_rate
}

// ---------------------------------------------------------------------------
extern "C" void kernel_launch(void* const* d_in, const int* in_sizes, int n_in,
                              void* d_out, int out_size, void* d_ws, size_t ws_size,
                              hipStream_t stream) {
  const float* x     = (const float*)d_in[0];  // [4,4096,4096]
  const float* Wg    = (const float*)d_in[1];  // [16,4096]
  const float* Wn    = (const float*)d_in[2];  // [16,4096]
  const float* noise = (const float*)d_in[3];  // [4,4096,16]

  float* out     = (float*)d_out;
  float* out_gw  = out;                                  // 16384*16
  float* out_idx = out_gw + (size_t)NTOK * NEXP;         // 16384*2
  float* out_tw  = out_idx + (size_t)NTOK * 2;           // 16384*2
  float* out5    = out_tw  + (size_t)NTOK * 2;           // 5 scalars

  float*    g_usage  = (float*)d_ws;
  unsigned* g_counts = (unsigned*)(g_usage + NEXP);

  moe_gate_zero<<<1, 32, 0, stream>>>(g_usage, g_counts);

  const int blocks = NTOK / (WAVES_PER_BLOCK * TOK_PER_WAVE);  // 256
  moe_gate_main<<<blocks, WAVES_PER_BLOCK * 32, 0, stream>>>(
      x, Wg, Wn, noise, out_gw, out_idx, out_tw, g_usage, g_counts);

  moe_gate_finalize<<<1, 32, 0, stream>>>(g_usage, g_counts, out5);
}